// GranuleCellLayer_35424890258157
// MI455X (gfx1250) — compile-verified
//
#include <hip/hip_runtime.h>
#include <hip/hip_bf16.h>

typedef __attribute__((ext_vector_type(2))) float v2f;
typedef __attribute__((ext_vector_type(8))) float v8f;

#define N_MOSSY   4096
#define N_GRANULE 16384
#define KTOP      491

// Async global->LDS copy, 16B per lane. LDS destination address comes from lds_off per lane.
__device__ __forceinline__ void async_g2l_b128(unsigned lds_off, const void* gaddr) {
  asm volatile("global_load_async_to_lds_b128 %0, %1, off"
               :: "v"(lds_off), "v"((unsigned long long)gaddr) : "memory");
}
__device__ __forceinline__ void wait_asynccnt0() {
  asm volatile("s_wait_asynccnt 0x0" ::: "memory");
}

// ---------------------------------------------------------------------------
// Main bandwidth kernel: partial g_exc over an m-slice.
//   gx[sIdx*N_GRANULE + g] = sum_{m in slice} W[g,m] * eff[m,g] * spikes[m]
// One wave computes 16 granule rows via v_wmma_f32_16x16x4_f32 accumulation.
// Staging uses gfx1250 async direct-to-LDS loads (ASYNCcnt).
// blockIdx.x in [0,128): granule tile;  blockIdx.y in [0,split): m slice.
// ---------------------------------------------------------------------------
__global__ __launch_bounds__(256) void granule_gemv(
    const float* __restrict__ spikes,   // [4096]
    const float* __restrict__ eff,      // [4096, 16384] row-major (m, g)
    const float* __restrict__ W,        // [16384, 4096] row-major (g, m)
    float* __restrict__ gx,             // [split, 16384] partial conductances
    const int split)
{
  __shared__ float sBlk[32];            // spike chunk (shared by all 8 waves)
  __shared__ float Wt[8][16][36];       // per-wave W tile, padded (conflict-free)
  __shared__ float Et[8][32][16];       // per-wave eff tile (transposed access)

  const int tid  = threadIdx.x;
  const int wave = tid >> 5;
  const int lane = tid & 31;
  const int g0   = blockIdx.x * 128 + wave * 16;
  const bool hiHalf = (lane >= 16);
  const int row  = lane & 15;

  const int mPer   = N_MOSSY / split;
  const int mBegin = blockIdx.y * mPer;
  const int mEnd   = mBegin + mPer;

  v8f c = {0.f, 0.f, 0.f, 0.f, 0.f, 0.f, 0.f, 0.f};

  for (int m0 = mBegin; m0 < mEnd; m0 += 32) {
    if (tid < 32) sBlk[tid] = spikes[m0 + tid];
    __syncthreads();

    // Stage W[g0..g0+15][m0..m0+31] via async-to-LDS: 128 float4 slots;
    // slot s -> r=s/8, q=s%8; consecutive lanes cover 128B per row (coalesced).
    #pragma unroll
    for (int j = 0; j < 4; ++j) {
      const int s = lane + 32 * j;
      const int r = s >> 3, q = s & 7;
      async_g2l_b128((unsigned)(size_t)&Wt[wave][r][q * 4],
                     W + (size_t)(g0 + r) * N_MOSSY + m0 + q * 4);
    }
    // Stage eff[m0+k][g0..g0+15]; rows with spike==0 are skipped (EXEC-masked
    // async op never fetches those 64B lines) and zero-filled instead.
    #pragma unroll
    for (int j = 0; j < 4; ++j) {
      const int k = (lane >> 2) + 8 * j;
      const int q = lane & 3;
      if (sBlk[k] != 0.0f) {
        async_g2l_b128((unsigned)(size_t)&Et[wave][k][q * 4],
                       eff + (size_t)(m0 + k) * N_GRANULE + g0 + q * 4);
      } else {
        *(float4*)&Et[wave][k][q * 4] = make_float4(0.f, 0.f, 0.f, 0.f);
      }
    }
    wait_asynccnt0();     // this wave's tiles landed in LDS
    __syncthreads();      // sBlk + everyone's zero-fill visible

    // 8 WMMAs cover the 32-wide m chunk, K=4 each.
    #pragma unroll
    for (int t = 0; t < 8; ++t) {
      const int kb = 4 * t;
      // Uniform skip when all 4 spikes in this K-group are zero (~41% of groups).
      const float gsum = sBlk[kb] + sBlk[kb + 1] + sBlk[kb + 2] + sBlk[kb + 3];
      if (gsum != 0.0f) {
        // A (16x4 f32): lanes 0-15 hold K={kb,kb+1}, lanes 16-31 hold K={kb+2,kb+3}
        const int k0 = kb + (hiHalf ? 2 : 0);
        const int k1 = k0 + 1;
        v2f a, b;
        a.x = Wt[wave][row][k0] * Et[wave][k0][row];
        a.y = Wt[wave][row][k1] * Et[wave][k1][row];
        // B (4x16 f32): row K broadcast across all 16 columns = spikes[K]
        b.x = sBlk[k0];
        b.y = sBlk[k1];
        c = __builtin_amdgcn_wmma_f32_16x16x4_f32(
            /*neg_a=*/false, a, /*neg_b=*/false, b,
            /*c_mod=*/(short)0, c, /*reuse_a=*/false, /*reuse_b=*/false);
      }
    }
    __syncthreads();
  }

  // D layout: vgpr v -> (M=v, N=lane) for lanes 0-15; (M=v+8, N=lane-16) for 16-31.
  // Every column N holds the same value; extract column 0.
  float* gxp = gx + (size_t)blockIdx.y * N_GRANULE;
  if (lane == 0) {
    #pragma unroll
    for (int v = 0; v < 8; ++v) gxp[g0 + v] = c[v];
  } else if (lane == 16) {
    #pragma unroll
    for (int v = 0; v < 8; ++v) gxp[g0 + 8 + v] = c[v];
  }
}

// ---------------------------------------------------------------------------
// Epilogue: reduce m-split partials (fixed order => deterministic), then
// LIF threshold + winner-take-all top-K. Single block, all in LDS.
// ---------------------------------------------------------------------------
__global__ __launch_bounds__(1024) void granule_epilogue(
    const float* __restrict__ gx, float* __restrict__ out, const int split)
{
  __shared__ int s_cnt;
  __shared__ int s_scan[1024];
  const int t = threadIdx.x;
  const int base = t * 16;

  unsigned key[16];
  int mycnt = 0;
  #pragma unroll
  for (int i = 0; i < 16; ++i) {
    float g = 0.0f;
    for (int s = 0; s < split; ++s)          // ascending order: deterministic
      g += gx[(size_t)s * N_GRANULE + base + i];
    // v = v_reset + (dt/tau)*((E_L - v0) + g*(E_E - v0)) = -65 + 0.2*(65*g)
    const float vm = -65.0f + 0.2f * (g * 65.0f);
    const bool sp = (vm >= -50.0f);
    // g_exc >= 0 always, so (bits | 0x80000000) is order-preserving.
    key[i] = sp ? (__float_as_uint(g) | 0x80000000u) : 0u;
    mycnt += sp ? 1 : 0;
  }

  if (t == 0) s_cnt = 0;
  __syncthreads();
  if (mycnt) atomicAdd(&s_cnt, mycnt);
  __syncthreads();
  const int nsp = s_cnt;

  if (nsp <= KTOP) {
    #pragma unroll
    for (int i = 0; i < 16; ++i) out[base + i] = (key[i] != 0u) ? 1.0f : 0.0f;
    return;
  }
  __syncthreads();

  // Binary search for V = K-th largest key: smallest T with count(key > T) < K.
  unsigned lo = 0u, hi = 0xFFFFFFFFu;
  while (lo < hi) {
    const unsigned mid = lo + ((hi - lo) >> 1);
    if (t == 0) s_cnt = 0;
    __syncthreads();
    int cloc = 0;
    #pragma unroll
    for (int i = 0; i < 16; ++i) cloc += (key[i] > mid) ? 1 : 0;
    if (cloc) atomicAdd(&s_cnt, cloc);
    __syncthreads();
    const int cg = s_cnt;
    if (cg < KTOP) hi = mid; else lo = mid + 1;
    __syncthreads();
  }
  const unsigned V = lo;

  // need = K - count(key > V); break ties at V by ascending index (top_k order).
  if (t == 0) s_cnt = 0;
  __syncthreads();
  int cgt = 0, ties = 0;
  #pragma unroll
  for (int i = 0; i < 16; ++i) {
    cgt  += (key[i] > V) ? 1 : 0;
    ties += (key[i] == V) ? 1 : 0;
  }
  if (cgt) atomicAdd(&s_cnt, cgt);
  __syncthreads();
  const int need = KTOP - s_cnt;
  __syncthreads();

  // Block-wide inclusive scan of tie counts -> global tie rank (index-ordered,
  // since thread t owns contiguous indices [16t, 16t+16)).
  s_scan[t] = ties;
  __syncthreads();
  for (int off = 1; off < 1024; off <<= 1) {
    const int x = (t >= off) ? s_scan[t - off] : 0;
    __syncthreads();
    s_scan[t] += x;
    __syncthreads();
  }
  int rank = s_scan[t] - ties;  // exclusive prefix

  #pragma unroll
  for (int i = 0; i < 16; ++i) {
    bool o = false;
    if (key[i] > V) {
      o = true;
    } else if (key[i] == V) {
      o = (rank < need);
      ++rank;
    }
    out[base + i] = o ? 1.0f : 0.0f;
  }
}

extern "C" void kernel_launch(void* const* d_in, const int* in_sizes, int n_in,
                              void* d_out, int out_size, void* d_ws, size_t ws_size,
                              hipStream_t stream) {
  (void)in_sizes; (void)n_in; (void)out_size;
  const float* spikes = (const float*)d_in[0];   // [4096]
  const float* eff    = (const float*)d_in[1];   // [4096, 16384]
  const float* W      = (const float*)d_in[2];   // [16384, 4096]
  float* out = (float*)d_out;                    // [16384] float 0/1
  float* gx  = (float*)d_ws;                     // [split, 16384] partials

  // m-dimension split factor: more concurrent waves -> more loads in flight.
  // Needs split*64KB of workspace; derived from ws_size (deterministic).
  int split = 1;
  while (split < 8 && (size_t)(split * 2) * (N_GRANULE * sizeof(float)) <= ws_size)
    split *= 2;

  granule_gemv<<<dim3(128, split), dim3(256), 0, stream>>>(spikes, eff, W, gx, split);
  granule_epilogue<<<dim3(1), dim3(1024), 0, stream>>>(gx, out, split);
}